// DifferentiableRasterization_2044404433137
// MI455X (gfx1250) — compile-verified
//
#include <hip/hip_runtime.h>

typedef __attribute__((ext_vector_type(2))) float v2f;
typedef __attribute__((ext_vector_type(8))) float v8f;

#define RSIZE   256
#define NVERT   8
#define NPIX    (RSIZE * RSIZE)   // 65536 pixels per polygon
#define NTILES  (NPIX / 16)       // 4096 16-pixel tiles per polygon pair
#define GRID_X  64
#define WAVES_X (GRID_X * 8)      // 512 wave slots striding the tiles (4096 = 512*8)

// One wave handles one polygon PAIR (2 polys x 8 verts = 16 WMMA M-slots) and a
// strided set of 16-pixel tiles. Per tile: one V_WMMA_F32_16X16X4_F32 produces
// all 256 squared distances (16 vertex slots x 16 pixels); each lane then owns
// one (poly, pixel) output: 8 in-lane sqrts for the distance sum, an even-odd
// inside test done entirely in the IEEE sign-bit domain (2 FMA + ~3 bit ops
// per edge, zero compares / selects / masks), sigmoid, coalesced store.
__global__ __launch_bounds__(256) void
diff_raster_wmma_kernel(const float* __restrict__ g_polys, float* __restrict__ g_out)
{
    __shared__ float s_raw[2 * NVERT * 2];   // this block's pair, raw [poly][vert][coord]

    const int tid  = threadIdx.x;
    const int lane = tid & 31;
    const int wave = tid >> 5;
    const int pair = blockIdx.y;             // 0..31 -> polys {2*pair, 2*pair+1}

    // ---- Stage the pair's 32 floats (128 B) into LDS via the gfx1250 async
    // path. Every wave issues its own (identical) copy and waits on its own
    // ASYNCcnt, so no workgroup barrier is needed.
    {
        unsigned lds_dst = (unsigned)(unsigned long long)&s_raw[lane]; // low 32b = LDS offset
        unsigned gofs    = (unsigned)((pair * 32 + lane) * sizeof(float));
        asm volatile("global_load_async_to_lds_b32 %0, %1, %2"
                     :: "v"(lds_dst), "v"(gofs), "s"(g_polys) : "memory");
        asm volatile("s_wait_asynccnt 0" ::: "memory");
    }

    const int half = lane >> 4;   // which poly of the pair this lane outputs for
    const int m    = lane & 15;   // A-slot index == B/C column (pixel) index

    // Reference transform: polys = roll(raw, 1, axis=-1) * 256
    //   -> x = raw[...,1]*256 , y = raw[...,0]*256
    // A matrix (16 vertex slots x K=4), row features [1, -2vx, -2vy, vx^2+vy^2]
    //   lanes 0-15 : v0=K0, v1=K1 of slot m ; lanes 16-31 : v0=K2, v1=K3 of slot m
    v2f A;
    {
        float vy = s_raw[m * 2 + 0] * 256.f;
        float vx = s_raw[m * 2 + 1] * 256.f;
        A.x = half ? (-2.f * vy) : 1.f;
        A.y = half ? (vx * vx + vy * vy) : (-2.f * vx);
    }

    // Hoist this lane's own polygon into registers (hot loop = regs only).
    float    ey[NVERT];                    // vertex y (tested against fj)
    float    dxe[NVERT], dye[NVERT], ce[NVERT];
    unsigned udye[NVERT];                  // dye bits (sign compared with tt)
    {
        float ex[NVERT];
#pragma unroll
        for (int e = 0; e < NVERT; ++e) {
            ey[e] = s_raw[(half * NVERT + e) * 2 + 0] * 256.f;
            ex[e] = s_raw[(half * NVERT + e) * 2 + 1] * 256.f;
        }
#pragma unroll
        for (int e = 0; e < NVERT; ++e) {
            dxe[e]  = ex[(e + 1) & 7] - ex[e];
            dye[e]  = ey[(e + 1) & 7] - ey[e];
            // invariant part of the cross product:
            // tt = dxe*fj - dye*fi + (dye*x - dxe*y)
            ce[e]   = dye[e] * ex[e] - dxe[e] * ey[e];
            udye[e] = __float_as_uint(dye[e]);
        }
    }

    const int wgid = blockIdx.x * 8 + wave;  // 0..511

    for (int t = wgid; t < NTILES; t += WAVES_X) {
        // Tiles never straddle an image row (16 | 256): row is wave-uniform.
        const float fi = (float)(t >> 4);                 // pts[:,0] (row)
        const float fj = (float)(((t & 15) << 4) + m);    // pts[:,1] (col)

        // B matrix (K=4 x 16 pixels), column features [fi^2+fj^2, fi, fj, 1]
        //   lanes 0-15 : v0=K0, v1=K1 of pixel m ; lanes 16-31 : v0=K2, v1=K3
        v2f B;
        B.x = half ? fj : (fi * fi + fj * fj);
        B.y = half ? 1.f : fi;

        v8f c = {};
        c = __builtin_amdgcn_wmma_f32_16x16x4_f32(false, A, false, B,
                                                  (short)0, c, false, false);

        // C layout: VGPR r, this lane -> d^2(vertex slot r + 8*half, pixel m):
        // exactly the 8 vertices of this lane's polygon -> in-lane reduction.
        float D = __builtin_amdgcn_sqrtf(c[0]);
#pragma unroll
        for (int r = 1; r < 8; ++r) D += __builtin_amdgcn_sqrtf(c[r]);

        // Even-odd inside test, entirely in the sign-bit domain (bit 31):
        //   (y > fj)              == signbit(fj - y)          (exact, +0 on ties)
        //   cond                  == bit31 of u(fj-y) ^ u(fj-yn)
        //   (tt>0) == (dye>0)     == bit31 of ~(u(tt) ^ u(dye))
        //   parity               ^= cond & sgn   -> fuses into v_bitop3
        float sj[NVERT];
#pragma unroll
        for (int v = 0; v < NVERT; ++v) sj[v] = fj - ey[v];

        unsigned par = 0u;
#pragma unroll
        for (int e = 0; e < NVERT; ++e) {
            const float    tt   = dxe[e] * fj + (ce[e] - dye[e] * fi); // 2 FMA
            const unsigned cond = __float_as_uint(sj[e]) ^ __float_as_uint(sj[(e + 1) & 7]);
            const unsigned nsgn = __float_as_uint(tt) ^ udye[e]; // bit31: signs differ
            par ^= cond & ~nsgn;
        }

        // z = C*D/512 with C=+1 inside (par bit31 set), -1 outside: XOR the
        // parity complement straight into the IEEE sign bit (no cmp/select).
        const unsigned flip = (~par) & 0x80000000u;
        const float    zmag = D * (1.f / (2.f * RSIZE));
        const float    z    = __uint_as_float(__float_as_uint(zmag) ^ flip);
        const float    s    = __builtin_amdgcn_rcpf(
                                 1.f + __builtin_amdgcn_exp2f(-z * 1.4426950408889634f));

        const int poly = pair * 2 + half;
        g_out[poly * NPIX + ((t << 4) + m)] = s;  // half-wave: 16 contiguous floats
    }
}

extern "C" void kernel_launch(void* const* d_in, const int* in_sizes, int n_in,
                              void* d_out, int out_size, void* d_ws, size_t ws_size,
                              hipStream_t stream) {
    (void)in_sizes; (void)n_in; (void)out_size; (void)d_ws; (void)ws_size;
    const float* polys = (const float*)d_in[0];   // (4,16,8,2) f32, flat 1024
    float*       out   = (float*)d_out;           // (4,16,256,256) f32

    dim3 grid(GRID_X, 32, 1);   // x: pixel-tile chunks, y: polygon pairs
    dim3 block(256, 1, 1);      // 8 waves of 32 (wave32)
    diff_raster_wmma_kernel<<<grid, block, 0, stream>>>(polys, out);
}